// Net_VIF_23888608101301
// MI455X (gfx1250) — compile-verified
//
#include <hip/hip_runtime.h>

// ---------------------------------------------------------------------------
// Net_VIF: 6 chained 4-D (quadrilinear) LUT interpolations.
//   B=8, H=W=512, DIM=17.  LUTs lut8,lut00..lut03: [4,17^4], lutpgf: [3,17^4].
// Strategy (MI455X):
//   * repack each LUT to [17^4][4] in d_ws -> each corner = one
//     global_load_b128 with compile-time immediate offset (L2-resident, 8 MB
//     total inside the 192 MB L2; HBM traffic is just 56 MB streaming)
//   * stage the 3 vi channels per block with the Tensor Data Mover
//     (tensor_load_to_lds, 2D tile 512x3, stride HW) and the ir channel with
//     global_load_async_to_lds_b32 (ASYNCcnt path)
//   * 2 pixels per thread -> 32 independent B128 gathers in flight per stage
//     to hide L2 latency across the serially-dependent 6-stage chain
// ---------------------------------------------------------------------------

constexpr int kD     = 17;
constexpr int kStr1  = 17;
constexpr int kStr2  = 17 * 17;
constexpr int kStr3  = 17 * 17 * 17;
constexpr int kNIdx  = 17 * 17 * 17 * 17;  // 83521
constexpr int kHW    = 512 * 512;          // 2^18
constexpr int kB     = 8;
constexpr int kNPix  = kB * kHW;           // 2,097,152
constexpr int kBlock = 256;
constexpr int kPPT   = 2;                  // pixels per thread
constexpr int kTile  = kBlock * kPPT;      // 512 pixels per block

#if __has_builtin(__builtin_amdgcn_tensor_load_to_lds)
#define HAVE_TDM 1
typedef unsigned int tdm_v4u __attribute__((ext_vector_type(4)));
typedef int          tdm_v8i __attribute__((ext_vector_type(8)));
typedef int          tdm_v4i __attribute__((ext_vector_type(4)));
#endif

__device__ __forceinline__ float satf(float v) {
  return fminf(fmaxf(v, 0.0f), 1.0f);
}

__device__ __forceinline__ void qsetup(const float4& in, int& base,
                                       float* wA, float* wB) {
  float a0 = satf(in.x) * (float)(kD - 1);
  float a1 = satf(in.y) * (float)(kD - 1);
  float a2 = satf(in.z) * (float)(kD - 1);
  float a3 = satf(in.w) * (float)(kD - 1);
  int i0 = (int)a0; if (i0 > kD - 2) i0 = kD - 2;
  int i1 = (int)a1; if (i1 > kD - 2) i1 = kD - 2;
  int i2 = (int)a2; if (i2 > kD - 2) i2 = kD - 2;
  int i3 = (int)a3; if (i3 > kD - 2) i3 = kD - 2;
  float f0 = a0 - (float)i0, f1 = a1 - (float)i1;
  float f2 = a2 - (float)i2, f3 = a3 - (float)i3;
  float g0 = 1.0f - f0, g1 = 1.0f - f1, g2 = 1.0f - f2, g3 = 1.0f - f3;
  wA[0] = g0 * g1; wA[1] = f0 * g1; wA[2] = g0 * f1; wA[3] = f0 * f1;
  wB[0] = g2 * g3; wB[1] = f2 * g3; wB[2] = g2 * f3; wB[3] = f2 * f3;
  base = ((i0 * kD + i1) * kD + i2) * kD + i3;
}

// Packed layout: [17^4][4] -> one B128 per corner, immediate offsets.
__device__ __forceinline__ float4 qinterp_packed(const float* __restrict__ lut,
                                                 const float4& in) {
  int base; float wA[4], wB[4];
  qsetup(in, base, wA, wB);
  const float4* p = reinterpret_cast<const float4*>(lut) + base;
  float4 acc = make_float4(0.f, 0.f, 0.f, 0.f);
#pragma unroll
  for (int c = 0; c < 16; ++c) {
    const int off = (c & 1) * kStr3 + ((c >> 1) & 1) * kStr2 +
                    ((c >> 2) & 1) * kStr1 + ((c >> 3) & 1);
    const float4 v = p[off];
    const float  w = wA[c & 3] * wB[(c >> 2) & 3];
    acc.x = fmaf(w, v.x, acc.x);
    acc.y = fmaf(w, v.y, acc.y);
    acc.z = fmaf(w, v.z, acc.z);
    acc.w = fmaf(w, v.w, acc.w);
  }
  return acc;
}

// Fallback: original [C][17^4] layout, scalar gathers.
template <int NC>
__device__ __forceinline__ float4 qinterp_unpacked(const float* __restrict__ lut,
                                                   const float4& in) {
  int base; float wA[4], wB[4];
  qsetup(in, base, wA, wB);
  const float* p = lut + base;
  float ax = 0.f, ay = 0.f, az = 0.f, aw = 0.f;
#pragma unroll
  for (int c = 0; c < 16; ++c) {
    const int off = (c & 1) * kStr3 + ((c >> 1) & 1) * kStr2 +
                    ((c >> 2) & 1) * kStr1 + ((c >> 3) & 1);
    const float w = wA[c & 3] * wB[(c >> 2) & 3];
    ax = fmaf(w, p[off], ax);
    ay = fmaf(w, p[kNIdx + off], ay);
    az = fmaf(w, p[2 * kNIdx + off], az);
    if (NC > 3) aw = fmaf(w, p[3 * kNIdx + off], aw);
  }
  return make_float4(ax, ay, az, aw);
}

__global__ void __launch_bounds__(kBlock)
lut_repack_kernel(const float* __restrict__ src, float* __restrict__ dst, int nc) {
  int i = blockIdx.x * kBlock + threadIdx.x;
  if (i >= kNIdx) return;
  float4 v;
  v.x = src[i];
  v.y = src[kNIdx + i];
  v.z = src[2 * kNIdx + i];
  v.w = (nc > 3) ? src[3 * kNIdx + i] : 0.0f;
  reinterpret_cast<float4*>(dst)[i] = v;
}

// Main fused kernel, packed LUTs, 2 px/thread, TDM + async input staging.
__global__ void __launch_bounds__(kBlock)
vif_fuse_packed(const float* __restrict__ vi, const float* __restrict__ ir,
                const float* __restrict__ l8, const float* __restrict__ l00,
                const float* __restrict__ l01, const float* __restrict__ l02,
                const float* __restrict__ l03, const float* __restrict__ lpg,
                float* __restrict__ out) {
  __shared__ float s_ch[4][kTile];   // channel-planar staging (8 KB)

  const int tid     = threadIdx.x;
  const int pixBase = blockIdx.x * kTile;        // uniform; kTile | kHW
  const int b       = pixBase >> 18;             // kHW == 2^18
  const int pos     = pixBase & (kHW - 1);

  // ---- stage vi channels (2D tile kTile x 3, row stride kHW) ----
  const float* viTile = vi + (size_t)b * 3 * kHW + pos;
#ifdef HAVE_TDM
  if ((tid >> 5) == 0) {             // one wave issues the DMA (EXEC ignored)
    const unsigned           ldsA = (unsigned)(size_t)&s_ch[0][0];
    const unsigned long long ga   = (unsigned long long)(size_t)viTile;
    tdm_v4u g0;
    g0[0] = 1u;                                           // count=1
    g0[1] = ldsA;                                         // lds_addr
    g0[2] = (unsigned)(ga & 0xffffffffu);                 // global_addr[31:0]
    g0[3] = (unsigned)((ga >> 32) & 0x01ffffffu)          // global_addr[56:32]
            | (2u << 30);                                 // type = 2 (image)
    tdm_v8i g1;
    g1[0] = (int)(2u << 16);                              // data_size = 4 B
    g1[1] = (int)(((unsigned)kHW & 0xffffu) << 16);       // tensor_dim0 lo16
    g1[2] = (int)(((unsigned)kHW >> 16) | (3u << 16));    // dim0 hi16 | dim1 lo16
    g1[3] = (int)((unsigned)kTile << 16);                 // tile_dim0
    g1[4] = 3;                                            // tile_dim1 = 3 rows
    g1[5] = kHW;                                          // dim0_stride lo32
    g1[6] = 0;
    g1[7] = 0;
    tdm_v4i gz4 = {0, 0, 0, 0};                           // groups 2/3 unused (2D)
    tdm_v8i gz8 = {0, 0, 0, 0, 0, 0, 0, 0};
    // clang-23 lane: 6-arg form (g0, g1, g2, g3, extra, cpol)
    __builtin_amdgcn_tensor_load_to_lds(g0, g1, gz4, gz4, gz8, 0);
    __builtin_amdgcn_s_wait_tensorcnt(0);
  }
#else
  {
    for (int c = 0; c < 3; ++c) {
      const float* src = viTile + (size_t)c * kHW;
      unsigned ldsC = (unsigned)(size_t)&s_ch[c][0];
#pragma unroll
      for (int j = 0; j < kPPT; ++j) {
        unsigned voff = (unsigned)(tid + j * kBlock) * 4u;
        asm volatile("global_load_async_to_lds_b32 %0, %1, %2"
                     :: "v"(ldsC + voff), "v"(voff), "s"(src) : "memory");
      }
    }
  }
#endif

  // ---- stage ir channel via async global->LDS (ASYNCcnt path) ----
  {
    const float* cir  = ir + (size_t)b * kHW + pos;
    unsigned     ldsI = (unsigned)(size_t)&s_ch[3][0];
#pragma unroll
    for (int j = 0; j < kPPT; ++j) {
      unsigned voff = (unsigned)(tid + j * kBlock) * 4u;
      asm volatile("global_load_async_to_lds_b32 %0, %1, %2"
                   :: "v"(ldsI + voff), "v"(voff), "s"(cir) : "memory");
    }
#if __has_builtin(__builtin_amdgcn_s_wait_asynccnt)
    __builtin_amdgcn_s_wait_asynccnt(0);
#else
    asm volatile("s_wait_asynccnt 0" ::: "memory");
#endif
  }
  __syncthreads();

  const int pA = tid;               // pixel A within tile
  const int pB = tid + kBlock;      // pixel B within tile
  float4 tA = make_float4(s_ch[0][pA], s_ch[1][pA], s_ch[2][pA], s_ch[3][pA]);
  float4 tB = make_float4(s_ch[0][pB], s_ch[1][pB], s_ch[2][pB], s_ch[3][pB]);

  // two independent dependency chains -> 32 B128 gathers in flight per stage
  tA = qinterp_packed(l8, tA);   tB = qinterp_packed(l8, tB);
  tA = qinterp_packed(l00, tA);  tB = qinterp_packed(l00, tB);
  tA = qinterp_packed(l01, tA);  tB = qinterp_packed(l01, tB);
  tA = qinterp_packed(l02, tA);  tB = qinterp_packed(l02, tB);
  tA = qinterp_packed(l03, tA);  tB = qinterp_packed(l03, tB);
  tA.x = satf(tA.x); tA.y = satf(tA.y); tA.z = satf(tA.z); tA.w = satf(tA.w);
  tB.x = satf(tB.x); tB.y = satf(tB.y); tB.z = satf(tB.z); tB.w = satf(tB.w);
  float4 oA = qinterp_packed(lpg, tA);
  float4 oB = qinterp_packed(lpg, tB);

  // out: [8,3,512,512]; streaming -> non-temporal stores
  const size_t oBase = (size_t)b * 3 * kHW + pos;
  __builtin_nontemporal_store(oA.x, &out[oBase + pA]);
  __builtin_nontemporal_store(oA.y, &out[oBase + pA + kHW]);
  __builtin_nontemporal_store(oA.z, &out[oBase + pA + 2 * (size_t)kHW]);
  __builtin_nontemporal_store(oB.x, &out[oBase + pB]);
  __builtin_nontemporal_store(oB.y, &out[oBase + pB + kHW]);
  __builtin_nontemporal_store(oB.z, &out[oBase + pB + 2 * (size_t)kHW]);
}

// Fallback when workspace is too small: direct loads, original LUT layout.
__global__ void __launch_bounds__(kBlock)
vif_fuse_unpacked(const float* __restrict__ vi, const float* __restrict__ ir,
                  const float* __restrict__ l8, const float* __restrict__ l00,
                  const float* __restrict__ l01, const float* __restrict__ l02,
                  const float* __restrict__ l03, const float* __restrict__ lpg,
                  float* __restrict__ out) {
  const int pix = blockIdx.x * kBlock + threadIdx.x;
  const int b   = pix >> 18;
  const int pos = pix & (kHW - 1);
  float4 x = make_float4(vi[((size_t)b * 3 + 0) * kHW + pos],
                         vi[((size_t)b * 3 + 1) * kHW + pos],
                         vi[((size_t)b * 3 + 2) * kHW + pos],
                         ir[(size_t)b * kHW + pos]);
  float4 t = qinterp_unpacked<4>(l8, x);
  t = qinterp_unpacked<4>(l00, t);
  t = qinterp_unpacked<4>(l01, t);
  t = qinterp_unpacked<4>(l02, t);
  t = qinterp_unpacked<4>(l03, t);
  t.x = satf(t.x); t.y = satf(t.y); t.z = satf(t.z); t.w = satf(t.w);
  float4 o = qinterp_unpacked<3>(lpg, t);
  const size_t oBase = (size_t)b * 3 * kHW + pos;
  out[oBase]                    = o.x;
  out[oBase + kHW]              = o.y;
  out[oBase + 2 * (size_t)kHW]  = o.z;
}

extern "C" void kernel_launch(void* const* d_in, const int* in_sizes, int n_in,
                              void* d_out, int out_size, void* d_ws, size_t ws_size,
                              hipStream_t stream) {
  (void)in_sizes; (void)n_in; (void)out_size;
  const float* vi    = (const float*)d_in[0];
  const float* ir    = (const float*)d_in[1];
  const float* lut8  = (const float*)d_in[2];
  const float* lut00 = (const float*)d_in[3];
  const float* lut01 = (const float*)d_in[4];
  const float* lut02 = (const float*)d_in[5];
  const float* lut03 = (const float*)d_in[6];
  const float* lutpg = (const float*)d_in[7];
  float* out = (float*)d_out;

  const size_t lutFloats = (size_t)kNIdx * 4;
  const size_t lutBytes  = lutFloats * sizeof(float);   // ~1.34 MB

  if (ws_size >= 6 * lutBytes) {
    float* w = (float*)d_ws;
    const float* srcs[6] = {lut8, lut00, lut01, lut02, lut03, lutpg};
    const int    ncs[6]  = {4, 4, 4, 4, 4, 3};
    const int rgrid = (kNIdx + kBlock - 1) / kBlock;
    for (int i = 0; i < 6; ++i)
      lut_repack_kernel<<<rgrid, kBlock, 0, stream>>>(srcs[i], w + i * lutFloats,
                                                      ncs[i]);
    vif_fuse_packed<<<kNPix / kTile, kBlock, 0, stream>>>(
        vi, ir, w, w + lutFloats, w + 2 * lutFloats, w + 3 * lutFloats,
        w + 4 * lutFloats, w + 5 * lutFloats, out);
  } else {
    vif_fuse_unpacked<<<kNPix / kBlock, kBlock, 0, stream>>>(
        vi, ir, lut8, lut00, lut01, lut02, lut03, lutpg, out);
  }
}